// AdaptiveDCA_9577777070135
// MI455X (gfx1250) — compile-verified
//
#include <hip/hip_runtime.h>
#include <hip/hip_bf16.h>

typedef float v2f __attribute__((ext_vector_type(2)));
typedef float v4f __attribute__((ext_vector_type(4)));
typedef float v8f __attribute__((ext_vector_type(8)));

#define TILE_M 128
#define TILE_N 128
#define TILE_K 32
#define LDA 36    // padded LDS row stride for A (floats): 36*m mod 64 distinct -> conflict-free b64 frag reads; rows 16B aligned
#define LDB 144   // padded LDS row stride for B (floats): rows 16B aligned; half-wave row shift = 16 banks -> conflict-free

// Async global->LDS DMA (CDNA5, GVS addressing: saddr base + 32-bit byte voffset).
// VDST = LDS byte address (low 32 bits of generic shared pointer == group-segment
// offset per the aperture rule). Tracked by ASYNCcnt.
__device__ __forceinline__ void async_copy_b128(unsigned lds_off, const void* sbase,
                                                int byte_off) {
    asm volatile("global_load_async_to_lds_b128 %0, %1, %2"
                 :: "v"(lds_off), "v"(byte_off), "s"(sbase)
                 : "memory");
}
__device__ __forceinline__ void wait_async0() {
    asm volatile("s_wait_asynccnt 0" ::: "memory");
}

// -----------------------------------------------------------------------------
// Kernel 1: qkv[b][o][p] = sum_c w_qkv[o][c] * x[b][c][p]
// Per batch: A = w_qkv (1536 x 512 row-major), B = x[b] (512 x 4096 row-major).
// 256 threads = 8 waves; wave wv computes rows [m0+16*wv, +16) x all 128 cols.
// Double-buffered LDS; staging via GLOBAL_LOAD_ASYNC_TO_LDS_B128 overlapped with
// the V_WMMA_F32_16X16X4_F32 compute of the current chunk.
// Fragment layouts per CDNA5 ISA 7.12.2:
//   A 16x4 : lane L (half=L>>4, m=L&15): VGPRs = A[m][k+2*half], A[m][k+2*half+1]
//   B 4x16 : lane L (n=L&15):            VGPRs = B[k+2*half][n], B[k+2*half+1][n]
//   C/D    : lane L, vgpr v -> row (8*half + v), col (L&15)
// -----------------------------------------------------------------------------
__global__ __launch_bounds__(256)
void qkv_gemm_kernel(const float* __restrict__ x, const float* __restrict__ w,
                     float* __restrict__ qkv)
{
    __shared__ float lds_a[2][TILE_M * LDA];
    __shared__ float lds_b[2][TILE_K * LDB];

    const int t  = threadIdx.x;
    const int m0 = blockIdx.x * TILE_M;     // 12 blocks -> 1536
    const int n0 = blockIdx.y * TILE_N;     // 32 blocks -> 4096
    const int bb = blockIdx.z;              // 8 batches

    const float* __restrict__ xb   = x   + (size_t)bb * 512  * 4096;
    float*       __restrict__ outb = qkv + (size_t)bb * 1536 * 4096;

    const int wv   = t >> 5;
    const int lane = t & 31;
    const int half = lane >> 4;
    const int ml   = lane & 15;

    // Per-thread staging assignments (constant across chunks).
    // A tile: 128 rows x 32 cols = 1024 float4; 4 per thread.
    int a_row[4], a_c4[4];
    int b_row[4], b_c4[4];
    #pragma unroll
    for (int j = 0; j < 4; ++j) {
        int idx  = j * 256 + t;
        a_row[j] = idx >> 3;               // 0..127
        a_c4[j]  = (idx & 7) << 2;         // 0,4,...,28
        b_row[j] = idx >> 5;               // 0..31
        b_c4[j]  = (idx & 31) << 2;        // 0,4,...,124
    }

    // Issue async stage of a K-chunk into buffer `buf`.
    auto stage = [&](int buf, int k0) {
        #pragma unroll
        for (int j = 0; j < 4; ++j) {
            unsigned la = (unsigned)(size_t)&lds_a[buf][a_row[j] * LDA + a_c4[j]];
            int      ga = (int)(((size_t)(m0 + a_row[j]) * 512 + (k0 + a_c4[j])) * 4);
            async_copy_b128(la, (const void*)w, ga);
        }
        #pragma unroll
        for (int j = 0; j < 4; ++j) {
            unsigned lb = (unsigned)(size_t)&lds_b[buf][b_row[j] * LDB + b_c4[j]];
            int      gb = (int)(((size_t)(k0 + b_row[j]) * 4096 + (n0 + b_c4[j])) * 4);
            async_copy_b128(lb, (const void*)xb, gb);
        }
    };

    v8f acc[8] = {};

    // Prologue: stage chunk 0 into buffer 0.
    stage(0, 0);
    wait_async0();
    __syncthreads();

    int cur = 0;
    for (int k0 = 0; k0 < 512; k0 += TILE_K) {
        // Prefetch next chunk into the other buffer (its readers all passed the
        // previous end-of-iteration barrier, so no WAR hazard).
        if (k0 + TILE_K < 512) stage(cur ^ 1, k0 + TILE_K);

        // Compute current chunk: 8 K-steps of 4, 8 N-tiles each.
        #pragma unroll
        for (int kk = 0; kk < TILE_K; kk += 4) {
            v2f a = *(const v2f*)(&lds_a[cur][(wv * 16 + ml) * LDA + kk + 2 * half]);
            #pragma unroll
            for (int j = 0; j < 8; ++j) {
                v2f b;
                b.x = lds_b[cur][(kk + 2 * half)     * LDB + j * 16 + ml];
                b.y = lds_b[cur][(kk + 2 * half + 1) * LDB + j * 16 + ml];
                acc[j] = __builtin_amdgcn_wmma_f32_16x16x4_f32(
                    /*neg_a=*/false, a, /*neg_b=*/false, b,
                    /*c_mod=*/(short)0, acc[j],
                    /*reuse_a=*/false, /*reuse_b=*/false);
            }
        }

        // Next chunk's DMA (issued before compute) must be visible to all waves.
        wait_async0();
        __syncthreads();
        cur ^= 1;
    }

    // Write back: row = m0 + wv*16 + 8*half + v, col = n0 + j*16 + ml
    #pragma unroll
    for (int j = 0; j < 8; ++j) {
        #pragma unroll
        for (int v = 0; v < 8; ++v) {
            outb[(size_t)(m0 + wv * 16 + 8 * half + v) * 4096 + (n0 + j * 16 + ml)]
                = acc[j][v];
        }
    }
}

// -----------------------------------------------------------------------------
// Kernel 2: pooled[b][c] = mean over 4096 pixels of x[b][c][:]
// -----------------------------------------------------------------------------
__global__ __launch_bounds__(256)
void pool_kernel(const float* __restrict__ x, float* __restrict__ pooled)
{
    __shared__ float red[256];
    const int bc = blockIdx.x;                       // b*512 + c (4096 blocks)
    const float* __restrict__ p = x + (size_t)bc * 4096;
    float s = 0.f;
    for (int i = threadIdx.x; i < 4096; i += 256) s += p[i];
    red[threadIdx.x] = s;
    __syncthreads();
    for (int off = 128; off > 0; off >>= 1) {
        if (threadIdx.x < off) red[threadIdx.x] += red[threadIdx.x + off];
        __syncthreads();
    }
    if (threadIdx.x == 0) pooled[bc] = red[0] * (1.0f / 4096.0f);
}

// -----------------------------------------------------------------------------
// Kernel 3: gate[b][i] = softmax_i(pooled[b] . w_gate[i] + b_gate[i])
// -----------------------------------------------------------------------------
__global__ __launch_bounds__(128)
void gate_kernel(const float* __restrict__ pooled, const float* __restrict__ w_gate,
                 const float* __restrict__ b_gate, float* __restrict__ gate)
{
    __shared__ float red[4 * 128];
    const int b   = blockIdx.x;
    const int tid = threadIdx.x;
    float acc[4] = {0.f, 0.f, 0.f, 0.f};
    for (int c = tid; c < 512; c += 128) {
        float pv = pooled[b * 512 + c];
        #pragma unroll
        for (int i = 0; i < 4; ++i) acc[i] += pv * w_gate[i * 512 + c];
    }
    #pragma unroll
    for (int i = 0; i < 4; ++i) red[i * 128 + tid] = acc[i];
    __syncthreads();
    for (int off = 64; off > 0; off >>= 1) {
        if (tid < off) {
            #pragma unroll
            for (int i = 0; i < 4; ++i)
                red[i * 128 + tid] += red[i * 128 + tid + off];
        }
        __syncthreads();
    }
    if (tid == 0) {
        float lg[4], ex[4];
        float mx = -1e30f;
        #pragma unroll
        for (int i = 0; i < 4; ++i) { lg[i] = red[i * 128] + b_gate[i]; mx = fmaxf(mx, lg[i]); }
        float s = 0.f;
        #pragma unroll
        for (int i = 0; i < 4; ++i) { ex[i] = __expf(lg[i] - mx); s += ex[i]; }
        float inv = 1.f / s;
        #pragma unroll
        for (int i = 0; i < 4; ++i) gate[b * 4 + i] = ex[i] * inv;
    }
}

// -----------------------------------------------------------------------------
// Kernel 4: dilated 3x3 attention per (pixel, head), gate folded into probs.
// Zero-padded neighbors -> logit exactly 0 and v contribution 0 (matches unfold).
// threadIdx.x = w (coalesced), blockDim = (64,4); grid = (16, B*8).
// -----------------------------------------------------------------------------
__global__ __launch_bounds__(256)
void attn_kernel(const float* __restrict__ qkv, const float* __restrict__ gate,
                 float* __restrict__ out)
{
    const int w  = threadIdx.x;
    const int h  = blockIdx.x * blockDim.y + threadIdx.y;
    const int bz = blockIdx.y;        // b*8 + head
    const int b  = bz >> 3;
    const int hh = bz & 7;
    const int g  = hh >> 1;           // dilation group
    const int th = hh & 1;            // head within group
    const int dil = 1 << g;

    const int HW = 4096;
    const size_t base = (size_t)b * 1536 * HW;
    const int chq = g * 128 + th * 64;
    const float* __restrict__ Q = qkv + base + (size_t)chq          * HW;
    const float* __restrict__ K = qkv + base + (size_t)(512 + chq)  * HW;
    const float* __restrict__ V = qkv + base + (size_t)(1024 + chq) * HW;

    int  noff[9];
    bool nval[9];
    #pragma unroll
    for (int i = 0; i < 3; ++i) {
        #pragma unroll
        for (int j = 0; j < 3; ++j) {
            int hi = h + (i - 1) * dil;
            int wj = w + (j - 1) * dil;
            bool v = (hi >= 0) && (hi < 64) && (wj >= 0) && (wj < 64);
            nval[i * 3 + j] = v;
            noff[i * 3 + j] = v ? (hi * 64 + wj) : 0;
        }
    }
    const int pix = h * 64 + w;

    float logit[9];
    #pragma unroll
    for (int e = 0; e < 9; ++e) logit[e] = 0.f;

    // pass 1: q.k logits; prefetch V rows for pass 2 (global_prefetch_b8)
    for (int c = 0; c < 64; ++c) {
        const float qc = Q[(size_t)c * HW + pix];
        const float* __restrict__ Kc = K + (size_t)c * HW;
        __builtin_prefetch(V + (size_t)c * HW + pix, 0, 1);
        #pragma unroll
        for (int e = 0; e < 9; ++e)
            if (nval[e]) logit[e] += qc * Kc[noff[e]];
    }

    // softmax over 9 (scale = 64^-0.5 = 0.125); padded logits stay exactly 0
    float mx = -1e30f;
    #pragma unroll
    for (int e = 0; e < 9; ++e) { logit[e] *= 0.125f; mx = fmaxf(mx, logit[e]); }
    float prob[9];
    float sum = 0.f;
    #pragma unroll
    for (int e = 0; e < 9; ++e) { prob[e] = __expf(logit[e] - mx); sum += prob[e]; }
    const float gmul = gate[b * 4 + g] / sum;   // fold gate with normalization
    #pragma unroll
    for (int e = 0; e < 9; ++e) prob[e] *= gmul;

    // pass 2: weighted v sum
    for (int c = 0; c < 64; ++c) {
        const float* __restrict__ Vc = V + (size_t)c * HW;
        float acc = 0.f;
        #pragma unroll
        for (int e = 0; e < 9; ++e)
            if (nval[e]) acc += prob[e] * Vc[noff[e]];
        out[((size_t)b * 512 + chq + c) * HW + pix] = acc;
    }
}

// -----------------------------------------------------------------------------
extern "C" void kernel_launch(void* const* d_in, const int* in_sizes, int n_in,
                              void* d_out, int out_size, void* d_ws, size_t ws_size,
                              hipStream_t stream)
{
    (void)in_sizes; (void)n_in; (void)out_size; (void)ws_size;

    const float* x      = (const float*)d_in[0];   // (8, 512, 64, 64)
    const float* w_qkv  = (const float*)d_in[1];   // (1536, 512)
    const float* w_gate = (const float*)d_in[2];   // (4, 512)
    const float* b_gate = (const float*)d_in[3];   // (4,)
    float* out = (float*)d_out;                    // (8, 512, 64, 64)

    float* qkv    = (float*)d_ws;                             // 8*1536*4096 floats (192 MB)
    float* pooled = qkv + (size_t)8 * 1536 * 4096;            // 8*512
    float* gate   = pooled + 8 * 512;                         // 8*4

    qkv_gemm_kernel<<<dim3(12, 32, 8), 256, 0, stream>>>(x, w_qkv, qkv);
    pool_kernel    <<<dim3(4096),      256, 0, stream>>>(x, pooled);
    gate_kernel    <<<dim3(8),         128, 0, stream>>>(pooled, w_gate, b_gate, gate);
    attn_kernel    <<<dim3(16, 64), dim3(64, 4), 0, stream>>>(qkv, gate, out);
}